// PatchNorm_36773509988971
// MI455X (gfx1250) — compile-verified
//
#include <hip/hip_runtime.h>
#include <stdint.h>

// PatchNorm: memory-bound gather+normalize.
//   patches: (B,S,D) f32      pos_c/h/w: (B,S) i32     pad: (B,S) bool(u8)
//   median,b: (C,H,W,D) f32   n: (C,H,W) f32           out: (B,S,D) f32
// Layout: one token per 64-thread block (2 wave32s); each lane owns one float4
// of the D=256 channel dim. Per-token scalars are wave-uniform -> SALU/s_load.

typedef float f4 __attribute__((ext_vector_type(4)));

#define POS_H 32
#define POS_W 32
#define DQ    64   // float4 chunks per token (D=256 / 4)

__global__ __launch_bounds__(64) void patchnorm_kernel(
    const float*   __restrict__ patches,
    const int*     __restrict__ pos_c,
    const int*     __restrict__ pos_h,
    const int*     __restrict__ pos_w,
    const uint8_t* __restrict__ pad,
    const float*   __restrict__ median,
    const float*   __restrict__ bmat,
    const float*   __restrict__ nmat,
    float*         __restrict__ out)
{
    const int token = blockIdx.x;          // wave-uniform (0 .. B*S-1)
    const int q     = threadIdx.x;         // 0..63, one float4 per lane

    // Per-token gather: uniform across the wave -> scalar loads.
    const int c    = pos_c[token];
    const int h    = pos_h[token];
    const int w    = pos_w[token];
    const int cell = (c * POS_H + h) * POS_W + w;
    const float ng = nmat[cell];
    const bool kill = (pad[token] != 0) | (ng <= 2.0f);

    const long pb = (long)token * DQ + q;  // float4 index into patches/out
    const long gb = (long)cell  * DQ + q;  // float4 index into median/b

    // Streamed once: non-temporal so the 128MB stream doesn't evict the
    // L2-resident gather tables (192MB L2).
    f4 p = __builtin_nontemporal_load((const f4*)patches + pb);
    // Gathered tables: regular temporal policy, stay hot in L2.
    f4 m = ((const f4*)median)[gb];
    f4 s = ((const f4*)bmat)[gb];

    f4 x;
#pragma unroll
    for (int k = 0; k < 4; ++k) {
        float stdv = s[k] * 1.4142135623730951f + 1e-6f;
        float v    = (p[k] - m[k]) / stdv;
        v          = fminf(fmaxf(v, -6.0f), 6.0f);
        x[k]       = kill ? 0.0f : v;   // clip(0)=0, so order matches reference
    }
    __builtin_nontemporal_store(x, (f4*)out + pb);
}

extern "C" void kernel_launch(void* const* d_in, const int* in_sizes, int n_in,
                              void* d_out, int out_size, void* d_ws, size_t ws_size,
                              hipStream_t stream) {
    const float*   patches = (const float*)d_in[0];
    const int*     pos_c   = (const int*)d_in[1];
    const int*     pos_h   = (const int*)d_in[2];
    const int*     pos_w   = (const int*)d_in[3];
    const uint8_t* pad     = (const uint8_t*)d_in[4];   // jnp.bool_ = 1 byte
    const float*   median  = (const float*)d_in[5];
    const float*   bmat    = (const float*)d_in[6];
    const float*   nmat    = (const float*)d_in[7];
    float*         out     = (float*)d_out;

    const int tokens = in_sizes[1];        // B*S (pos_channels element count)

    patchnorm_kernel<<<tokens, 64, 0, stream>>>(
        patches, pos_c, pos_h, pos_w, pad, median, bmat, nmat, out);
}